// MatrixConfigurationTrainer_27161373180469
// MI455X (gfx1250) — compile-verified
//
#include <hip/hip_runtime.h>
#include <hip/hip_bf16.h>

#define NDIM 64
#define DFEAT 16
#define BATCH 8192
#define BT 16          // batch points per workgroup
#define ITERS 128      // shifted power-iteration steps

typedef float v2f __attribute__((ext_vector_type(2)));
typedef float v8f __attribute__((ext_vector_type(8)));

// workspace layout (float offsets)
#define WS_HEFF   0       // 4096 float2 (interleaved re,im) = 8192 floats
#define WS_COMM   8192    // 120 per-pair commutator norms
#define WS_RECON  8320    // 512 per-WG recon partial sums
#define WS_QF     8832    // 512 per-WG qf partial sums

// ---------------- Heff = 0.5 * sum_k A_k^2 ----------------
__global__ __launch_bounds__(256) void heff_kernel(const float2* __restrict__ A,
                                                   float2* __restrict__ Heff) {
  int idx = blockIdx.x * 256 + threadIdx.x;   // 0..4095
  int m = idx >> 6, n = idx & 63;
  float ar = 0.f, ai = 0.f;
  for (int k = 0; k < DFEAT; ++k) {
    const float2* Ak = A + k * NDIM * NDIM;
    for (int l = 0; l < NDIM; ++l) {
      float2 a = Ak[m * NDIM + l];
      float2 b = Ak[l * NDIM + n];
      ar += a.x * b.x - a.y * b.y;
      ai += a.x * b.y + a.y * b.x;
    }
  }
  float2 h; h.x = 0.5f * ar; h.y = 0.5f * ai;
  Heff[idx] = h;
}

// ---------------- commutator norms: one block per (i<j) pair ----------------
__global__ __launch_bounds__(256) void comm_kernel(const float2* __restrict__ A,
                                                   float* __restrict__ commPart) {
  __shared__ float sred[256];
  int p = blockIdx.x;
  int i = 0, rem = p;
  while (rem >= DFEAT - 1 - i) { rem -= DFEAT - 1 - i; ++i; }
  int j = i + 1 + rem;
  const float2* Ai = A + i * NDIM * NDIM;
  const float2* Aj = A + j * NDIM * NDIM;
  float acc = 0.f;
  for (int e = threadIdx.x; e < NDIM * NDIM; e += 256) {
    int m = e >> 6, n = e & 63;
    float cr = 0.f, ci = 0.f;
    for (int l = 0; l < NDIM; ++l) {
      float2 a = Ai[m * NDIM + l], b = Aj[l * NDIM + n];
      cr += a.x * b.x - a.y * b.y;
      ci += a.x * b.y + a.y * b.x;
      float2 c = Aj[m * NDIM + l], d = Ai[l * NDIM + n];
      cr -= c.x * d.x - c.y * d.y;
      ci -= c.x * d.y + c.y * d.x;
    }
    acc += cr * cr + ci * ci;
  }
  sred[threadIdx.x] = acc;
  __syncthreads();
  for (int s = 128; s > 0; s >>= 1) {
    if (threadIdx.x < (unsigned)s) sred[threadIdx.x] += sred[threadIdx.x + s];
    __syncthreads();
  }
  if (threadIdx.x == 0) commPart[p] = sqrtf(sred[0]);
}

// ---------------- main solver: 16 points per workgroup ----------------
__device__ inline v8f wmma_f32(v2f a, v2f b, v8f c) {
  return __builtin_amdgcn_wmma_f32_16x16x4_f32(false, a, false, b, (short)0, c,
                                               false, false);
}

__global__ __launch_bounds__(256) void solve_kernel(const float* __restrict__ pts,
                                                    const float4* __restrict__ A4,
                                                    const float4* __restrict__ H4,
                                                    float* __restrict__ reconPart,
                                                    float* __restrict__ qfPart) {
  __shared__ float Vr[NDIM][BT], Vi[NDIM][BT];   // current vectors (cols = points)
  __shared__ float Yr[NDIM][BT], Yi[NDIM][BT];   // GEMM result planes
  __shared__ float xs[DFEAT][BT];                // column scales x[b][k]
  __shared__ float xloc[BT][DFEAT];
  __shared__ float shiftc[BT];                   // (sigma - 0.5|x|^2) per column
  __shared__ float red[16][BT], red2[16][BT];
  __shared__ float nrm[BT];
  __shared__ float eA[BT][DFEAT], eA2[BT][DFEAT];

  const int tid = threadIdx.x;
  const int b0 = blockIdx.x * BT;

  { // load points tile: xs[k][col] = points[b0+col][k]
    int col = tid >> 4, k = tid & 15;
    float x = pts[(b0 + col) * DFEAT + k];
    xloc[col][k] = x;
    xs[k][col] = x;
  }
  __syncthreads();
  if (tid < BT) {
    // sigma_b = 0.5 * sum_k (1+|x_k|)^2 + 1  >= lambda_max(H)  (||A_k||_2 <= 1)
    float sb = 0.f, x2 = 0.f;
    for (int k = 0; k < DFEAT; ++k) {
      float x = xloc[tid][k];
      float ax = fabsf(x);
      sb += 0.5f * (1.0f + ax) * (1.0f + ax);
      x2 += x * x;
    }
    shiftc[tid] = sb + 1.0f - 0.5f * x2;
  }
  { // init V with deterministic hash
    int col = tid & 15, rseg = tid >> 4;
    for (int q = 0; q < 4; ++q) {
      int r = rseg * 4 + q;
      unsigned h = (unsigned)(r * 2654435761u) ^ (unsigned)(col * 40503u + 0x9e3779b9u);
      Vr[r][col] = ((h >> 8) & 0xFFFF) * (1.0f / 65536.0f) - 0.5f;
      Vi[r][col] = ((h >> 16) & 0xFFFF) * (1.0f / 65536.0f) - 0.5f;
    }
  }
  __syncthreads();

  // wave -> (M-tile, plane) assignment; WMMA layouts per ISA 7.12.2:
  //   A 16x4 f32 : lane L holds (M = L%16, K = v + 2*(L/16)) for VGPR v
  //   C/D 16x16  : VGPR r -> row r + 8*(L/16), col L%16
  const int lane = tid & 31;
  const int wave = tid >> 5;
  const int mt = wave & 3;        // M tile (rows mt*16..+15)
  const int pp = wave >> 2;       // 0 = real plane, 1 = imag plane
  const int colB = lane & 15;     // fragment column == batch point in tile
  const int half = lane >> 4;
  const int rowA = mt * 16 + (lane & 15);
  const int colU = tid & 15, rsegU = tid >> 4;

  // wave-uniform plane selection hoisted out of all loops (no per-lane selects)
  const float (*BA)[BT] = (pp == 0) ? Vr : Vi;   // feeds A_re chain
  const float (*BB)[BT] = (pp == 0) ? Vi : Vr;   // feeds A_im chain
  const float sgnB = (pp == 0) ? -1.0f : 1.0f;
  float (*Yp)[BT] = (pp == 0) ? Yr : Yi;

  // per-point scales kept in registers; invariant across iterations
  float sreg[DFEAT];
#pragma unroll
  for (int j = 0; j < DFEAT; ++j) sreg[j] = xs[j][colB];

  const int fbase = (rowA * NDIM + 2 * half) >> 1;   // float4 index base in a matrix

  for (int it = 0; it < ITERS; ++it) {
    // Y = sum_k x_bk (A_k V) - Heff V   via f32 WMMA (complex as two planes)
    v8f acc = {};
    for (int k0 = 0; k0 < NDIM; k0 += 4) {
      const int kk = k0 + 2 * half;
      // V fragments: loaded once per k0, shared across all 17 matrices
      const float a0 = BA[kk][colB], a1 = BA[kk + 1][colB];
      const float b0 = BB[kk][colB] * sgnB, b1 = BB[kk + 1][colB] * sgnB;
      const int fidx = fbase + (k0 >> 1);
#pragma unroll
      for (int j = 0; j < DFEAT; ++j) {
        float4 m = A4[j * 2048 + fidx];          // one b128: (re,im) x 2 K-steps
        v2f ar; ar[0] = m.x; ar[1] = m.z;
        v2f ai; ai[0] = m.y; ai[1] = m.w;
        float s = sreg[j];
        v2f bA; bA[0] = a0 * s; bA[1] = a1 * s;
        v2f bB; bB[0] = b0 * s; bB[1] = b1 * s;
        acc = wmma_f32(ar, bA, acc);
        acc = wmma_f32(ai, bB, acc);
      }
      { // Heff term, weight -1 folded into B fragments
        float4 m = H4[fidx];
        v2f ar; ar[0] = m.x; ar[1] = m.z;
        v2f ai; ai[0] = m.y; ai[1] = m.w;
        v2f bA; bA[0] = -a0; bA[1] = -a1;
        v2f bB; bB[0] = -b0; bB[1] = -b1;
        acc = wmma_f32(ar, bA, acc);
        acc = wmma_f32(ai, bB, acc);
      }
    }
#pragma unroll
    for (int r = 0; r < 8; ++r)
      Yp[mt * 16 + r + 8 * half][colB] = acc[r];
    __syncthreads();
    { // y = shift*v + Y ; accumulate column norms
      float part = 0.f;
      float c = shiftc[colU];
#pragma unroll
      for (int q = 0; q < 4; ++q) {
        int r = rsegU * 4 + q;
        float yr = c * Vr[r][colU] + Yr[r][colU];
        float yi = c * Vi[r][colU] + Yi[r][colU];
        Yr[r][colU] = yr; Yi[r][colU] = yi;
        part += yr * yr + yi * yi;
      }
      red[rsegU][colU] = part;
    }
    __syncthreads();
    if (tid < BT) {
      float s = 0.f;
      for (int q = 0; q < 16; ++q) s += red[q][tid];
      nrm[tid] = rsqrtf(fmaxf(s, 1e-30f));
    }
    __syncthreads();
    {
      float sc = nrm[colU];
#pragma unroll
      for (int q = 0; q < 4; ++q) {
        int r = rsegU * 4 + q;
        Vr[r][colU] = Yr[r][colU] * sc;
        Vi[r][colU] = Yi[r][colU] * sc;
      }
    }
    __syncthreads();
  }

  // expectations: w = A_k v ; <A_k> = Re(v^H w) ; <A_k^2> = ||w||^2
  for (int k = 0; k < DFEAT; ++k) {
    v8f acc = {};
#pragma unroll
    for (int k0 = 0; k0 < NDIM; k0 += 4) {
      const int kk = k0 + 2 * half;
      const float a0 = BA[kk][colB], a1 = BA[kk + 1][colB];
      const float b0 = BB[kk][colB] * sgnB, b1 = BB[kk + 1][colB] * sgnB;
      float4 m = A4[k * 2048 + fbase + (k0 >> 1)];
      v2f ar; ar[0] = m.x; ar[1] = m.z;
      v2f ai; ai[0] = m.y; ai[1] = m.w;
      v2f bA; bA[0] = a0; bA[1] = a1;
      v2f bB; bB[0] = b0; bB[1] = b1;
      acc = wmma_f32(ar, bA, acc);
      acc = wmma_f32(ai, bB, acc);
    }
#pragma unroll
    for (int r = 0; r < 8; ++r)
      Yp[mt * 16 + r + 8 * half][colB] = acc[r];
    __syncthreads();
    {
      float pd = 0.f, ps = 0.f;
#pragma unroll
      for (int q = 0; q < 4; ++q) {
        int r = rsegU * 4 + q;
        float wr = Yr[r][colU], wi = Yi[r][colU];
        pd += Vr[r][colU] * wr + Vi[r][colU] * wi;
        ps += wr * wr + wi * wi;
      }
      red[rsegU][colU] = pd;
      red2[rsegU][colU] = ps;
    }
    __syncthreads();
    if (tid < BT) {
      float d = 0.f, s = 0.f;
      for (int q = 0; q < 16; ++q) { d += red[q][tid]; s += red2[q][tid]; }
      eA[tid][k] = d;
      eA2[tid][k] = s;
    }
    __syncthreads();
  }

  // per-point losses -> per-WG partials (deterministic, no atomics)
  if (tid < BT) {
    float recon = 0.f, qf = 0.f;
    for (int k = 0; k < DFEAT; ++k) {
      float ea = eA[tid][k];
      float dxe = xloc[tid][k] - ea;
      recon += dxe * dxe;
      qf += eA2[tid][k] - ea * ea;
    }
    red[0][tid] = recon;
    red2[0][tid] = qf;
  }
  __syncthreads();
  if (tid == 0) {
    float r = 0.f, q = 0.f;
    for (int c = 0; c < BT; ++c) { r += red[0][c]; q += red2[0][c]; }
    reconPart[blockIdx.x] = r;
    qfPart[blockIdx.x] = q;
  }
}

// ---------------- finalize ----------------
__global__ void finalize_kernel(const float* __restrict__ commPart,
                                const float* __restrict__ reconPart,
                                const float* __restrict__ qfPart,
                                float* __restrict__ out) {
  if (threadIdx.x == 0 && blockIdx.x == 0) {
    float comm = 0.f;
    for (int i = 0; i < (DFEAT * (DFEAT - 1)) / 2; ++i) comm += commPart[i];
    float r = 0.f, q = 0.f;
    for (int i = 0; i < BATCH / BT; ++i) { r += reconPart[i]; q += qfPart[i]; }
    r /= (float)BATCH;
    q /= (float)BATCH;
    out[0] = r + 0.1f + 0.1f * q;   // faithful to source: adds the WEIGHT scalar
    out[1] = r;
    out[2] = comm;
    out[3] = q;
  }
}

extern "C" void kernel_launch(void* const* d_in, const int* in_sizes, int n_in,
                              void* d_out, int out_size, void* d_ws, size_t ws_size,
                              hipStream_t stream) {
  (void)in_sizes; (void)n_in; (void)out_size; (void)ws_size;
  const float*  pts = (const float*)d_in[0];
  const float2* A   = (const float2*)d_in[1];   // complex64 interleaved
  float* ws = (float*)d_ws;
  float2* Heff      = (float2*)(ws + WS_HEFF);
  float* commPart   = ws + WS_COMM;
  float* reconPart  = ws + WS_RECON;
  float* qfPart     = ws + WS_QF;
  float* out        = (float*)d_out;

  heff_kernel<<<16, 256, 0, stream>>>(A, Heff);
  comm_kernel<<<(DFEAT * (DFEAT - 1)) / 2, 256, 0, stream>>>(A, commPart);
  solve_kernel<<<BATCH / BT, 256, 0, stream>>>(pts, (const float4*)A,
                                               (const float4*)Heff,
                                               reconPart, qfPart);
  finalize_kernel<<<1, 32, 0, stream>>>(commPart, reconPart, qfPart, out);
}